// DGraFormer_framework_37924561224032
// MI455X (gfx1250) — compile-verified
//
#include <hip/hip_runtime.h>
#include <hip/hip_bf16.h>

typedef __attribute__((ext_vector_type(2))) float v2f;
typedef __attribute__((ext_vector_type(8))) float v8f;

#define BB 32
#define NN 64
#define SS 336
#define DD 32
#define KLAY 3

// ---------------------------------------------------------------------------
// Pre-kernel: collapse (w_start, b_start, w_mlp, b_mlp) into 8 scalars.
//   out = c0*p0 + c1*p1 + c2*p2 + c3*p3 + d1*q1 + d2*q2 + d3*q3 + const
// with p_k = (A^T)^k x,  q_k = (A^T)^k 1.
// ---------------------------------------------------------------------------
__global__ void dgra_coef_kernel(const float* __restrict__ w_start,
                                 const float* __restrict__ b_start,
                                 const float* __restrict__ w_mlp,
                                 const float* __restrict__ b_mlp,
                                 float* __restrict__ coef) {
    if (threadIdx.x == 0 && blockIdx.x == 0) {
        float W[4], Ww[4], Wb[4];
        for (int k = 0; k < 4; ++k) {
            float sw = 0.f, sww = 0.f, swb = 0.f;
            for (int d = 0; d < DD; ++d) {
                float wm = w_mlp[k * DD + d];
                sw  += wm;
                sww += w_start[d] * wm;
                swb += b_start[d] * wm;
            }
            W[k] = sw; Ww[k] = sww; Wb[k] = swb;
        }
        const float beta = 0.05f, g = 0.95f;
        const float g2 = g * g, g3 = g2 * g;
        coef[0] = Ww[0] + beta * (W[1] + W[2] + W[3]);    // c0 (on p0 = x)
        coef[1] = g  * Ww[1] + g  * beta * (W[2] + W[3]); // c1
        coef[2] = g2 * Ww[2] + g2 * beta * W[3];          // c2
        coef[3] = g3 * Ww[3];                             // c3
        coef[4] = g  * Wb[1];                             // d1 (on q1)
        coef[5] = g2 * Wb[2];                             // d2
        coef[6] = g3 * Wb[3];                             // d3
        coef[7] = Wb[0] + b_mlp[0];                       // const
    }
}

// ---------------------------------------------------------------------------
// Main kernel: one workgroup per (b, s) tile.
// 128 threads = 4 waves; wave w owns output rows [16w, 16w+16).
// Per iteration: [p_k, q_k] = A^T [p_{k-1}, q_{k-1}] via V_WMMA_F32_16X16X4_F32,
// with B-matrix columns 0/1 = (p, q), columns 2..15 = 0 (branchless select).
// ---------------------------------------------------------------------------
__global__ __launch_bounds__(128)
void dgra_mp_kernel(const float* __restrict__ x,
                    const float* __restrict__ adj,
                    const float* __restrict__ coef,
                    float* __restrict__ out) {
    __shared__ alignas(16) float amat[NN * NN];      // adj[b,s] as [n][m], m fast
    __shared__ alignas(16) float pq[KLAY + 1][NN * 2]; // interleaved {p[n], q[n]}

    const int tile = blockIdx.x;          // 0 .. B*S-1
    const int b = tile / SS;
    const int s = tile - b * SS;
    const int t = threadIdx.x;

    // ---- stream the 16KB adj tile into LDS (coalesced b128 loads) ----
    const float* __restrict__ asrc = adj + (size_t)tile * (NN * NN);
#pragma unroll
    for (int i = 0; i < 8; ++i) {
        int idx = (i * 128 + t) * 4;
        *(float4*)&amat[idx] = *(const float4*)&asrc[idx];
    }
    // ---- p0 = x[b,:,s], q0 = ones (interleaved) ----
    if (t < NN) {
        pq[0][t * 2 + 0] = x[((size_t)b * NN + t) * SS + s];
        pq[0][t * 2 + 1] = 1.0f;
    }
    // ---- uniform scalar coefficients ----
    float cf[8];
#pragma unroll
    for (int i = 0; i < 8; ++i) cf[i] = coef[i];

    __syncthreads();

    const int wave = t >> 5;
    const int lane = t & 31;
    const int lj   = lane & 15;   // WMMA N column / M row-within-tile
    const int hi   = lane >> 4;   // half-wave select
    const int m0   = wave * 16;   // this wave's output-row tile

    // Branchless column-select multipliers: col 0 = p, col 1 = q, rest 0.
    const float selp = (lj == 0) ? 1.0f : 0.0f;
    const float selq = (lj == 1) ? 1.0f : 0.0f;

    for (int k = 1; k <= KLAY; ++k) {
        const float* __restrict__ pqin = pq[k - 1];
        v8f acc = {};   // C starts at zero, accumulate over 16 K-tiles
#pragma unroll
        for (int kt = 0; kt < 16; ++kt) {
            const int n0 = kt * 4 + hi * 2;
            // A-tile (16x4 f32): lane (hi,lj) holds A[m0+lj, K=hi*2 + {0,1}]
            //   A(m, k) = adj^T[m, n] = amat[n*64 + m]
            v2f av;
            av.x = amat[n0 * NN + m0 + lj];
            av.y = amat[(n0 + 1) * NN + m0 + lj];
            // B-tile (4x16 f32): lane (hi,lj) holds B[K=hi*2 + {0,1}, N=lj]
            // single b128 broadcast load: {p[n0], q[n0], p[n0+1], q[n0+1]}
            const float4 pv = *(const float4*)&pqin[n0 * 2];
            v2f bv;
            bv.x = fmaf(selp, pv.x, selq * pv.y);
            bv.y = fmaf(selp, pv.z, selq * pv.w);
            // D = A x B + C   (EXEC is all-ones here: uniform control flow)
            acc = __builtin_amdgcn_wmma_f32_16x16x4_f32(
                false, av, false, bv, (short)0, acc, false, false);
        }
        // D layout: VGPR r -> rows M=r (lanes 0-15) and M=8+r (lanes 16-31),
        // N = lane&15. Columns 0/1 carry the new p/q vectors.
        if (lj < 2) {
            const int mbase = m0 + hi * 8;
#pragma unroll
            for (int r = 0; r < 8; ++r) pq[k][(mbase + r) * 2 + lj] = acc[r];
        }
        __syncthreads();
    }

    // ---- combine: out[b,m,s] = c0*x + sum ck*pk + sum dk*qk + const ----
    if (t < NN) {
        float v = cf[7]
                + cf[0] * pq[0][t * 2 + 0]
                + cf[1] * pq[1][t * 2 + 0]
                + cf[2] * pq[2][t * 2 + 0]
                + cf[3] * pq[3][t * 2 + 0]
                + cf[4] * pq[1][t * 2 + 1]
                + cf[5] * pq[2][t * 2 + 1]
                + cf[6] * pq[3][t * 2 + 1];
        out[((size_t)b * NN + t) * SS + s] = v;
    }
}

extern "C" void kernel_launch(void* const* d_in, const int* in_sizes, int n_in,
                              void* d_out, int out_size, void* d_ws, size_t ws_size,
                              hipStream_t stream) {
    const float* x       = (const float*)d_in[0];   // [32,64,336]
    const float* adj     = (const float*)d_in[1];   // [32,336,64,64]
    const float* w_start = (const float*)d_in[2];   // [1,32]
    const float* b_start = (const float*)d_in[3];   // [32]
    const float* w_mlp   = (const float*)d_in[4];   // [128,1]
    const float* b_mlp   = (const float*)d_in[5];   // [1]
    float* out  = (float*)d_out;                    // [32,64,336]
    float* coef = (float*)d_ws;                     // 8 scalars

    dgra_coef_kernel<<<1, 32, 0, stream>>>(w_start, b_start, w_mlp, b_mlp, coef);
    dgra_mp_kernel<<<BB * SS, 128, 0, stream>>>(x, adj, coef, out);
}